// YieldGNN_27238682591805
// MI455X (gfx1250) — compile-verified
//
#include <hip/hip_runtime.h>
#include <hip/hip_bf16.h>

typedef __attribute__((ext_vector_type(2))) float v2f;
typedef __attribute__((ext_vector_type(8))) float v8f;

#define NN 100000
#define NE 1600000
#define HID 64
#define FIN 128
#define EPSV 1e-5f

// ---------------- degree / normalization ----------------
__global__ void k_init_deg(float* deg, int n) {
    int i = blockIdx.x * blockDim.x + threadIdx.x;
    if (i < n) deg[i] = 1.0f;  // self loop
}

__global__ void k_count_deg(const int* __restrict__ dst, float* __restrict__ deg, int e) {
    int i = blockIdx.x * blockDim.x + threadIdx.x;
    if (i < e) unsafeAtomicAdd(&deg[dst[i]], 1.0f);
}

__global__ void k_deg_to_dinv(float* deg, int n) {
    int i = blockIdx.x * blockDim.x + threadIdx.x;
    if (i < n) deg[i] = rsqrtf(deg[i]);
}

// ---------------- f32 WMMA GEMM: out[N,64] = (A[N,K] @ W[K,64]) * dinv[row] ----------------
// One block = 4 waves; wave w computes the 16x16 tile (rows blockIdx.x*16.., cols w*16..).
__global__ __launch_bounds__(128) void k_gemm_wmma(const float* __restrict__ A,
                                                   const float* __restrict__ W,
                                                   const float* __restrict__ dinv,
                                                   float* __restrict__ out,
                                                   int n, int K) {
    const int lane = threadIdx.x & 31;
    const int wv   = threadIdx.x >> 5;   // 0..3 -> column tile
    const int half = lane >> 4;          // 0 or 1
    const int l15  = lane & 15;
    int row = blockIdx.x * 16 + l15;
    if (row >= n) row = n - 1;           // clamp: keep EXEC all-1s for WMMA
    const int col = wv * 16 + l15;
    const float* Arow = A + (size_t)row * K;

    v8f c = {};
    for (int k = 0; k < K; k += 4) {
        const int kk = k + 2 * half;
        v2f a, b;
        // A 16x4 fragment: a[v] holds (M=l15, K=kk+v)
        a[0] = Arow[kk];
        a[1] = Arow[kk + 1];
        // B 4x16 fragment: b[v] holds (K=kk+v, N=col); W is [K,64] row-major
        b[0] = W[(size_t)kk * HID + col];
        b[1] = W[(size_t)(kk + 1) * HID + col];
        c = __builtin_amdgcn_wmma_f32_16x16x4_f32(false, a, false, b, (short)0, c, false, false);
    }
#pragma unroll
    for (int r = 0; r < 8; ++r) {
        int orow = blockIdx.x * 16 + r + 8 * half;  // C/D layout: vgpr r -> M=r (lanes<16), M=r+8
        if (orow < n) out[(size_t)orow * HID + col] = c[r] * dinv[orow];
    }
}

// ---------------- edge scatter: agg[dst] += hw'[src] ----------------
__global__ __launch_bounds__(256) void k_scatter(const int* __restrict__ src,
                                                 const int* __restrict__ dst,
                                                 const float* __restrict__ hw,
                                                 float* __restrict__ agg) {
    unsigned tid = blockIdx.x * blockDim.x + threadIdx.x;  // NE*16 < 2^31
    if (tid >= (unsigned)NE * 16u) return;
    const int e  = (int)(tid >> 4);
    const int f0 = (int)(tid & 15) * 4;
    const int s = src[e], d = dst[e];
    const float4 v = *(const float4*)(hw + (size_t)s * HID + f0);
    float* p = agg + (size_t)d * HID + f0;
    unsafeAtomicAdd(p + 0, v.x);
    unsafeAtomicAdd(p + 1, v.y);
    unsafeAtomicAdd(p + 2, v.z);
    unsafeAtomicAdd(p + 3, v.w);
}

// ---------------- batchnorm ----------------
__global__ void k_zero_stats(float* stats) {
    if (threadIdx.x < 128) stats[threadIdx.x] = 0.0f;
}

// y[i,f] = agg[i,f]*dinv[i] + b[f]; accumulate per-feature sum / sumsq
__global__ __launch_bounds__(256) void k_bn_stats(const float* __restrict__ agg,
                                                  const float* __restrict__ dinv,
                                                  const float* __restrict__ bias,
                                                  float* __restrict__ stats, int n) {
    __shared__ float s1[256];
    __shared__ float s2[256];
    const int f  = threadIdx.x & 63;
    const int rg = threadIdx.x >> 6;  // 0..3 rows per block iteration
    const float bf = bias[f];
    float sum = 0.0f, sq = 0.0f;
    for (int i = blockIdx.x * 4 + rg; i < n; i += gridDim.x * 4) {
        float y = agg[(size_t)i * HID + f] * dinv[i] + bf;
        sum += y;
        sq  += y * y;
    }
    s1[threadIdx.x] = sum;
    s2[threadIdx.x] = sq;
    __syncthreads();
    if (threadIdx.x < 64) {
        float a1 = s1[f] + s1[f + 64] + s1[f + 128] + s1[f + 192];
        float a2 = s2[f] + s2[f + 64] + s2[f + 128] + s2[f + 192];
        unsafeAtomicAdd(&stats[f], a1);
        unsafeAtomicAdd(&stats[64 + f], a2);
    }
}

__global__ void k_bn_finalize(const float* __restrict__ stats,
                              const float* __restrict__ g,
                              const float* __restrict__ bt,
                              float* __restrict__ ss, int n) {
    int f = threadIdx.x;
    if (f < 64) {
        float invn  = 1.0f / (float)n;
        float mu    = stats[f] * invn;
        float var   = stats[64 + f] * invn - mu * mu;  // biased var
        float scale = g[f] * rsqrtf(var + EPSV);
        ss[f]       = scale;
        ss[64 + f]  = bt[f] - mu * scale;
    }
}

// h[i,f] = relu((agg[i,f]*dinv[i] + b[f]) * scale[f] + shift[f])
__global__ __launch_bounds__(256) void k_bn_apply(const float* __restrict__ agg,
                                                  const float* __restrict__ dinv,
                                                  const float* __restrict__ bias,
                                                  const float* __restrict__ ss,
                                                  float* __restrict__ h, int n) {
    int tid = blockIdx.x * blockDim.x + threadIdx.x;
    if (tid >= n * HID) return;
    const int i = tid >> 6;
    const int f = tid & 63;
    float y = agg[tid] * dinv[i] + bias[f];
    float v = y * ss[f] + ss[64 + f];
    h[tid] = v > 0.0f ? v : 0.0f;
}

// ---------------- MLP head: out[i] = relu(h[i]@lw1+lb1) @ lw2 + lb2 ----------------
__global__ __launch_bounds__(256) void k_head(const float* __restrict__ h,
                                              const float* __restrict__ lw1,
                                              const float* __restrict__ lb1,
                                              const float* __restrict__ lw2,
                                              const float* __restrict__ lb2,
                                              float* __restrict__ out, int n) {
    int i = blockIdx.x * blockDim.x + threadIdx.x;
    if (i >= n) return;
    float acc[32];
#pragma unroll
    for (int j = 0; j < 32; ++j) acc[j] = lb1[j];
    const float* hr = h + (size_t)i * HID;
    for (int f = 0; f < HID; ++f) {
        float hv = hr[f];
#pragma unroll
        for (int j = 0; j < 32; ++j) acc[j] += hv * lw1[f * 32 + j];
    }
    float o = lb2[0];
#pragma unroll
    for (int j = 0; j < 32; ++j) {
        float z = acc[j] > 0.0f ? acc[j] : 0.0f;
        o += z * lw2[j];
    }
    out[i] = o;
}

// ---------------- launcher ----------------
extern "C" void kernel_launch(void* const* d_in, const int* in_sizes, int n_in,
                              void* d_out, int out_size, void* d_ws, size_t ws_size,
                              hipStream_t stream) {
    const float* x   = (const float*)d_in[0];
    const int*   ei  = (const int*)d_in[1];
    const int*   src = ei;            // edge_index[0]
    const int*   dst = ei + NE;       // edge_index[1]
    const float* Wm[3] = {(const float*)d_in[2], (const float*)d_in[6], (const float*)d_in[10]};
    const float* bm[3] = {(const float*)d_in[3], (const float*)d_in[7], (const float*)d_in[11]};
    const float* gm[3] = {(const float*)d_in[4], (const float*)d_in[8], (const float*)d_in[12]};
    const float* btm[3] = {(const float*)d_in[5], (const float*)d_in[9], (const float*)d_in[13]};
    const float* lw1 = (const float*)d_in[14];
    const float* lb1 = (const float*)d_in[15];
    const float* lw2 = (const float*)d_in[16];
    const float* lb2 = (const float*)d_in[17];

    // workspace carve-up (256B aligned)
    char* ws = (char*)d_ws;
    size_t off = 0;
    auto carve = [&](size_t bytes) {
        void* p = ws + off;
        off = (off + bytes + 255) & ~(size_t)255;
        return p;
    };
    float* dinv  = (float*)carve((size_t)NN * 4);        // degree -> dinv (in place)
    float* stats = (float*)carve(128 * 4);               // [sum(64), sumsq(64)]
    float* ss    = (float*)carve(128 * 4);               // [scale(64), shift(64)]
    float* B0    = (float*)carve((size_t)NN * HID * 4);  // hw' = (h@W)*dinv[row]
    float* B1    = (float*)carve((size_t)NN * HID * 4);  // agg_raw
    float* B2    = (float*)carve((size_t)NN * HID * 4);  // post BN+ReLU h

    // symmetric normalization factors
    k_init_deg<<<(NN + 255) / 256, 256, 0, stream>>>(dinv, NN);
    k_count_deg<<<(NE + 255) / 256, 256, 0, stream>>>(dst, dinv, NE);
    k_deg_to_dinv<<<(NN + 255) / 256, 256, 0, stream>>>(dinv, NN);

    const float* hin = x;
    int K = FIN;
    for (int L = 0; L < 3; ++L) {
        // hw' = (h @ W) * dinv[row]   (f32 WMMA)
        k_gemm_wmma<<<(NN + 15) / 16, 128, 0, stream>>>(hin, Wm[L], dinv, B0, NN, K);
        // self-loop contribution: agg_raw = hw'
        hipMemcpyAsync(B1, B0, (size_t)NN * HID * 4, hipMemcpyDeviceToDevice, stream);
        // edge scatter: agg_raw[dst] += hw'[src]
        k_scatter<<<((unsigned)NE * 16u + 255u) / 256u, 256, 0, stream>>>(src, dst, B0, B1);
        // batchnorm (training stats, biased var) + ReLU
        k_zero_stats<<<1, 128, 0, stream>>>(stats);
        k_bn_stats<<<1024, 256, 0, stream>>>(B1, dinv, bm[L], stats, NN);
        k_bn_finalize<<<1, 64, 0, stream>>>(stats, gm[L], btm[L], ss, NN);
        k_bn_apply<<<(NN * HID + 255) / 256, 256, 0, stream>>>(B1, dinv, bm[L], ss, B2, NN);
        hin = B2;
        K = HID;
    }
    k_head<<<(NN + 255) / 256, 256, 0, stream>>>(B2, lw1, lb1, lw2, lb2, (float*)d_out, NN);
}